// GAT_40888088657987
// MI455X (gfx1250) — compile-verified
//
#include <hip/hip_runtime.h>
#include <math.h>

typedef __attribute__((ext_vector_type(2))) float v2f;
typedef __attribute__((ext_vector_type(8))) float v8f;

// ---------------------------------------------------------------------------
// helpers
// ---------------------------------------------------------------------------
__device__ __forceinline__ int esrc(const int* ei, int E, int e) {
  return (e < E) ? ei[e] : (e - E);            // self loops appended after E
}
__device__ __forceinline__ int edst(const int* ei, int E, int e) {
  return (e < E) ? ei[E + e] : (e - E);
}

// float atomic max via sign-split int/uint atomics (valid with -inf init)
__device__ __forceinline__ void atomicMaxF(float* addr, float val) {
  if (val >= 0.0f) atomicMax((int*)addr, __float_as_int(val));
  else             atomicMin((unsigned int*)addr, __float_as_uint(val));
}

// ---------------------------------------------------------------------------
// fill
// ---------------------------------------------------------------------------
__global__ void fill_f32(float* __restrict__ p, float v, int n) {
  int i = blockIdx.x * blockDim.x + threadIdx.x;
  if (i < n) p[i] = v;
}

// ---------------------------------------------------------------------------
// C[MxNc] = A[MxK] @ B[KxNc] (+bias), row-major f32, via V_WMMA_F32_16X16X4_F32
// block = 128 threads (4 waves), wave w -> N-tile (blockIdx.y*4 + w)
// requires M%16==0, Nc%64==0, K%4==0
// ---------------------------------------------------------------------------
__global__ __launch_bounds__(128) void gemm_wmma_f32(
    const float* __restrict__ A, const float* __restrict__ B,
    const float* __restrict__ bias, float* __restrict__ C,
    int M, int Nc, int K) {
  const int wave = threadIdx.x >> 5;
  const int lane = threadIdx.x & 31;
  const int half = lane >> 4;    // 0: K pair {0,1}, 1: K pair {2,3}
  const int l16  = lane & 15;
  const int m0 = blockIdx.x * 16;
  const int n0 = (blockIdx.y * 4 + wave) * 16;
  if (m0 >= M || n0 >= Nc) return;

  const float* __restrict__ arow = A + (long long)(m0 + l16) * K + 2 * half;
  const float* __restrict__ bcol = B + (long long)(2 * half) * Nc + n0 + l16;

  v8f acc = {0.f, 0.f, 0.f, 0.f, 0.f, 0.f, 0.f, 0.f};
  for (int k = 0; k < K; k += 4) {
    float2 af = *(const float2*)(arow + k);          // A[m][k+2h], A[m][k+2h+1]
    v2f a = {af.x, af.y};
    v2f b = {bcol[(long long)k * Nc], bcol[(long long)(k + 1) * Nc]};
    acc = __builtin_amdgcn_wmma_f32_16x16x4_f32(
        /*neg_a=*/false, a, /*neg_b=*/false, b,
        /*c_mod=*/(short)0, acc, /*reuse_a=*/false, /*reuse_b=*/false);
  }
  float bs = bias ? bias[n0 + l16] : 0.0f;
#pragma unroll
  for (int r = 0; r < 8; ++r) {
    int m = m0 + half * 8 + r;                       // VGPR r -> row r / 8+r
    C[(long long)m * Nc + n0 + l16] = acc[r] + bs;
  }
}

// ---------------------------------------------------------------------------
// per-(node,head) attention logits:  al = sum_d h[n,h,d]*a[h,d]
// ---------------------------------------------------------------------------
template <int H, int D>
__global__ void att_logits(const float* __restrict__ hf,
                           const float* __restrict__ a_s,
                           const float* __restrict__ a_d,
                           float* __restrict__ als, float* __restrict__ ald,
                           int N) {
  int idx = blockIdx.x * blockDim.x + threadIdx.x;
  if (idx >= N * H) return;
  int h = idx & (H - 1);                               // H is a power of two
  const float* row = hf + (long long)idx * D;          // [N, H*D] contiguous
  const float* as = a_s + h * D;
  const float* ad = a_d + h * D;
  float s = 0.f, t = 0.f;
#pragma unroll 8
  for (int d = 0; d < D; ++d) {
    float v = row[d];
    s += v * as[d];
    t += v * ad[d];
  }
  als[idx] = s;
  ald[idx] = t;
}

// ---------------------------------------------------------------------------
// edge pass 1: leaky_relu logit, store, atomic segment max over dst
// ---------------------------------------------------------------------------
template <int H>
__global__ void edge_logit_max(const int* __restrict__ ei, int E, int Ep,
                               const float* __restrict__ als,
                               const float* __restrict__ ald,
                               float* __restrict__ elog, float* __restrict__ m) {
  int idx = blockIdx.x * blockDim.x + threadIdx.x;
  if (idx >= Ep * H) return;
  int e = idx / H, h = idx & (H - 1);                  // shifts: H = 8 or 1
  int s = esrc(ei, E, e), d = edst(ei, E, e);
  float v = als[s * H + h] + ald[d * H + h];
  v = (v > 0.f) ? v : 0.2f * v;                        // leaky_relu slope 0.2
  elog[idx] = v;
  atomicMaxF(&m[d * H + h], v);
}

// ---------------------------------------------------------------------------
// edge pass 2: ex = exp(e - m[dst]); store in-place; atomic segment sum
// ---------------------------------------------------------------------------
template <int H>
__global__ void edge_exp_sum(const int* __restrict__ ei, int E, int Ep,
                             float* __restrict__ elog,
                             const float* __restrict__ m,
                             float* __restrict__ den) {
  int idx = blockIdx.x * blockDim.x + threadIdx.x;
  if (idx >= Ep * H) return;
  int e = idx / H, h = idx & (H - 1);
  int d = edst(ei, E, e);
  float ex = expf(elog[idx] - m[d * H + h]);
  elog[idx] = ex;
  atomicAdd(&den[d * H + h], ex);
}

// ---------------------------------------------------------------------------
// edge pass 3: out[dst, f] += alpha * h[src, f]   (wave per edge, lane = d)
// alpha (incl. the divide) fully hoisted; gathers/atomics unrolled so the
// compiler can clause them. Lane f-pattern: f = lane + 32k -> coalesced.
// ---------------------------------------------------------------------------
template <int H, int D>
__global__ void edge_aggregate(const int* __restrict__ ei, int E, int Ep,
                               const float* __restrict__ hf,
                               const float* __restrict__ ex,
                               const float* __restrict__ den,
                               float* __restrict__ outp) {
  constexpr int F  = H * D;
  constexpr int CH = F / 32;                           // chunks per lane
  int wv = (blockIdx.x * blockDim.x + threadIdx.x) >> 5;
  int lane = threadIdx.x & 31;
  if (wv >= Ep) return;
  int s = esrc(ei, E, wv), d = edst(ei, E, wv);

  float alpha[CH];
#pragma unroll
  for (int k = 0; k < CH; ++k) {
    int h = (lane + 32 * k) / D;                       // compile-time shift
    alpha[k] = ex[(long long)wv * H + h] / (den[d * H + h] + 1e-16f);
  }
  const float* __restrict__ hrow = hf + (long long)s * F;
  float* __restrict__ orow = outp + (long long)d * F;
  float g[CH];
#pragma unroll
  for (int k = 0; k < CH; ++k) g[k] = hrow[lane + 32 * k];
#pragma unroll
  for (int k = 0; k < CH; ++k)
    atomicAdd(&orow[lane + 32 * k], alpha[k] * g[k]);
}

// ---------------------------------------------------------------------------
// h = elu(gat0 + b0 + r0)        (layer-0 epilogue, F = 256)
// ---------------------------------------------------------------------------
__global__ void post_layer0(const float* __restrict__ gat0,
                            const float* __restrict__ b0,
                            const float* __restrict__ r0,
                            float* __restrict__ h, int total) {
  int i = blockIdx.x * blockDim.x + threadIdx.x;
  if (i >= total) return;
  float v = gat0[i] + b0[i & 255] + r0[i];
  h[i] = (v > 0.f) ? v : expm1f(v);
}

// ---------------------------------------------------------------------------
// final: o = gat1 + b1 + r1; log_softmax over 64 cols (wave per node)
// ---------------------------------------------------------------------------
__global__ void final_logsoftmax(const float* __restrict__ gat1,
                                 const float* __restrict__ r1,
                                 const float* __restrict__ b1,
                                 float* __restrict__ out, int N) {
  int wv = (blockIdx.x * blockDim.x + threadIdx.x) >> 5;
  int lane = threadIdx.x & 31;
  if (wv >= N) return;
  long long base = (long long)wv * 64;
  float o0 = gat1[base + lane]      + b1[lane]      + r1[base + lane];
  float o1 = gat1[base + 32 + lane] + b1[32 + lane] + r1[base + 32 + lane];
  float mx = fmaxf(o0, o1);
#pragma unroll
  for (int off = 16; off > 0; off >>= 1) mx = fmaxf(mx, __shfl_xor(mx, off, 32));
  float s = expf(o0 - mx) + expf(o1 - mx);
#pragma unroll
  for (int off = 16; off > 0; off >>= 1) s += __shfl_xor(s, off, 32);
  float lse = mx + logf(s);
  out[base + lane] = o0 - lse;
  out[base + 32 + lane] = o1 - lse;
}

// ---------------------------------------------------------------------------
// host orchestration
// ---------------------------------------------------------------------------
static inline unsigned cdiv(long long a, long long b) { return (unsigned)((a + b - 1) / b); }

extern "C" void kernel_launch(void* const* d_in, const int* in_sizes, int n_in,
                              void* d_out, int out_size, void* d_ws, size_t ws_size,
                              hipStream_t stream) {
  (void)n_in; (void)out_size; (void)ws_size;
  const float* x     = (const float*)d_in[0];
  const int*   ei    = (const int*)d_in[1];
  const float* W0    = (const float*)d_in[2];
  const float* aS0   = (const float*)d_in[3];
  const float* aD0   = (const float*)d_in[4];
  const float* b0    = (const float*)d_in[5];
  const float* Wr0   = (const float*)d_in[6];
  const float* br0   = (const float*)d_in[7];
  const float* W1    = (const float*)d_in[8];
  const float* aS1   = (const float*)d_in[9];
  const float* aD1   = (const float*)d_in[10];
  const float* b1    = (const float*)d_in[11];
  const float* Wr1   = (const float*)d_in[12];
  const float* br1   = (const float*)d_in[13];

  const int N  = in_sizes[0] / 256;   // 50000
  const int E  = in_sizes[1] / 2;     // 800000
  const int Ep = E + N;               // + self loops

  // carve workspace
  float* ws = (float*)d_ws;
  float* h0   = ws; ws += (size_t)N * 256;
  float* r0   = ws; ws += (size_t)N * 256;
  float* gat0 = ws; ws += (size_t)N * 256;
  float* hbuf = ws; ws += (size_t)N * 256;
  float* als0 = ws; ws += (size_t)N * 8;
  float* ald0 = ws; ws += (size_t)N * 8;
  float* m0   = ws; ws += (size_t)N * 8;
  float* den0 = ws; ws += (size_t)N * 8;
  float* elog0= ws; ws += (size_t)Ep * 8;
  float* z1   = ws; ws += (size_t)N * 64;
  float* r1   = ws; ws += (size_t)N * 64;
  float* gat1 = ws; ws += (size_t)N * 64;
  float* als1 = ws; ws += (size_t)N;
  float* ald1 = ws; ws += (size_t)N;
  float* m1   = ws; ws += (size_t)N;
  float* den1 = ws; ws += (size_t)N;
  float* elog1= ws; ws += (size_t)Ep;

  const dim3 blk256(256);

  // ---- layer 0 dense (WMMA) ----
  {
    dim3 g(N / 16, 256 / 64), b(128);
    gemm_wmma_f32<<<g, b, 0, stream>>>(x, W0,  nullptr, h0, N, 256, 256);
    gemm_wmma_f32<<<g, b, 0, stream>>>(x, Wr0, br0,     r0, N, 256, 256);
  }
  att_logits<8, 32><<<cdiv((long long)N * 8, 256), blk256, 0, stream>>>(h0, aS0, aD0, als0, ald0, N);

  fill_f32<<<cdiv((long long)N * 8, 256), blk256, 0, stream>>>(m0,   -INFINITY, N * 8);
  fill_f32<<<cdiv((long long)N * 8, 256), blk256, 0, stream>>>(den0, 0.0f,      N * 8);
  fill_f32<<<cdiv((long long)N * 256, 256), blk256, 0, stream>>>(gat0, 0.0f, N * 256);

  edge_logit_max<8><<<cdiv((long long)Ep * 8, 256), blk256, 0, stream>>>(ei, E, Ep, als0, ald0, elog0, m0);
  edge_exp_sum<8>  <<<cdiv((long long)Ep * 8, 256), blk256, 0, stream>>>(ei, E, Ep, elog0, m0, den0);
  edge_aggregate<8, 32><<<cdiv(Ep, 8), blk256, 0, stream>>>(ei, E, Ep, h0, elog0, den0, gat0);

  post_layer0<<<cdiv((long long)N * 256, 256), blk256, 0, stream>>>(gat0, b0, r0, hbuf, N * 256);

  // ---- layer 1 dense (WMMA) ----
  {
    dim3 g(N / 16, 64 / 64), b(128);
    gemm_wmma_f32<<<g, b, 0, stream>>>(hbuf, W1,  nullptr, z1, N, 64, 256);
    gemm_wmma_f32<<<g, b, 0, stream>>>(hbuf, Wr1, br1,     r1, N, 64, 256);
  }
  att_logits<1, 64><<<cdiv(N, 256), blk256, 0, stream>>>(z1, aS1, aD1, als1, ald1, N);

  fill_f32<<<cdiv(N, 256), blk256, 0, stream>>>(m1,   -INFINITY, N);
  fill_f32<<<cdiv(N, 256), blk256, 0, stream>>>(den1, 0.0f,      N);
  fill_f32<<<cdiv((long long)N * 64, 256), blk256, 0, stream>>>(gat1, 0.0f, N * 64);

  edge_logit_max<1><<<cdiv(Ep, 256), blk256, 0, stream>>>(ei, E, Ep, als1, ald1, elog1, m1);
  edge_exp_sum<1>  <<<cdiv(Ep, 256), blk256, 0, stream>>>(ei, E, Ep, elog1, m1, den1);
  edge_aggregate<1, 64><<<cdiv(Ep, 8), blk256, 0, stream>>>(ei, E, Ep, z1, elog1, den1, gat1);

  final_logsoftmax<<<cdiv(N, 8), blk256, 0, stream>>>(gat1, r1, b1, (float*)d_out, N);
}